// MultiScaleP2RLoss_71451075936419
// MI455X (gfx1250) — compile-verified
//
#include <hip/hip_runtime.h>
#include <hip/hip_bf16.h>

typedef float v2f __attribute__((ext_vector_type(2)));
typedef float v8f __attribute__((ext_vector_type(8)));

// Reference-fixed shapes
constexpr int  Bimg = 16;
constexpr int  Himg = 768;
constexpr int  Wimg = 768;
constexpr int  Npts = 2048;
constexpr long HW   = (long)Himg * Wimg;          // 589824
constexpr int  TPB  = 256;                        // 8 waves of 32
constexpr int  ELEMS_PER_BLOCK = TPB * 8;         // 2048 floats (2 x float4 / thread)
constexpr int  BLOCKS_PER_IMG  = (int)(HW / ELEMS_PER_BLOCK); // 288, exact

// ---------------------------------------------------------------------------
// Full wave32 sum via V_WMMA_F32_16X16X4_F32.
// A (16x4 f32): VGPR0 = p (K=0 lanes 0-15, K=2 lanes 16-31), VGPR1 = 0.
// B = all ones  => D[m][*] = p[m] + p[m+16] (row sums, replicated over N).
// D layout: lane<16: VGPR v holds M=v; lane>=16: M=v+8.
// Sum of 8 D VGPRs -> half-wave sums; one shfl_xor(16) completes the total.
// Requires EXEC all ones: call sites are convergent, blocks are full.
// ---------------------------------------------------------------------------
__device__ __forceinline__ float wave_sum32(float p) {
  v2f a; a.x = p;   a.y = 0.0f;
  v2f b; b.x = 1.0f; b.y = 1.0f;
  v8f c = {};
  v8f d = __builtin_amdgcn_wmma_f32_16x16x4_f32(
      /*neg_a=*/false, a, /*neg_b=*/false, b,
      /*c_mod=*/(short)0, c, /*reuse_a=*/false, /*reuse_b=*/false);
  float s = d[0] + d[1] + d[2] + d[3] + d[4] + d[5] + d[6] + d[7];
  s += __shfl_xor(s, 16, 32);
  return s;  // full 32-lane sum in every lane
}

// Block (256 threads = 8 waves) sum; result valid in thread 0.
__device__ __forceinline__ float block_sum(float v, float* lds8) {
  float w = wave_sum32(v);
  const int lane = threadIdx.x & 31;
  const int wid  = threadIdx.x >> 5;
  if (lane == 0) lds8[wid] = w;
  __syncthreads();
  float r = 0.0f;
  if (threadIdx.x == 0) {
#pragma unroll
    for (int i = 0; i < TPB / 32; ++i) r += lds8[i];
  }
  return r;
}

// ---------------------------------------------------------------------------
// Kernel 0: zero the 3*B accumulator slots (S_pred, S_tgt, mse).
// ---------------------------------------------------------------------------
__global__ void k_init_acc(float* __restrict__ acc) {
  const int i = threadIdx.x;
  if (i < 3 * Bimg) acc[i] = 0.0f;
}

// ---------------------------------------------------------------------------
// Kernel 1: zero target + per-image sum of pred (streaming, float4).
// grid = (BLOCKS_PER_IMG, Bimg), block = 256.
// ---------------------------------------------------------------------------
__global__ void __launch_bounds__(TPB)
k_zero_and_predsum(const float* __restrict__ pred,
                   float* __restrict__ target,
                   float* __restrict__ sPred) {
  __shared__ float lds8[TPB / 32];
  const int  b    = blockIdx.y;
  const long base = (long)b * HW + (long)blockIdx.x * ELEMS_PER_BLOCK;
  const float4 z  = make_float4(0.f, 0.f, 0.f, 0.f);
  float p = 0.0f;
#pragma unroll
  for (int k = 0; k < 2; ++k) {
    const long e = base + 4L * (threadIdx.x + k * TPB);
    const float4 v = *(const float4*)(pred + e);
    *(float4*)(target + e) = z;
    p += (v.x + v.y) + (v.z + v.w);
  }
  const float s = block_sum(p, lds8);
  if (threadIdx.x == 0) atomicAdd(&sPred[b], s);
}

// ---------------------------------------------------------------------------
// Kernel 2: Gaussian splat scatter + per-image target-sum (sum of valid taps).
// grid = (Npts/TPB, Bimg) = (8, 16), block = 256 (one point per thread).
// ---------------------------------------------------------------------------
__global__ void __launch_bounds__(TPB)
k_scatter(const float* __restrict__ points,
          float* __restrict__ target,
          float* __restrict__ sTgt,
          const int* __restrict__ H_in_p,
          const int* __restrict__ W_in_p) {
  __shared__ float lds8[TPB / 32];
  const int b = blockIdx.y;
  const int n = blockIdx.x * TPB + threadIdx.x;  // < 2048, exact

  const float scale_w = (float)Wimg / (float)(*W_in_p);
  const float scale_h = (float)Himg / (float)(*H_in_p);

  const long  pbase = ((long)b * Npts + n) * 2;
  const float px = points[pbase + 0];
  const float py = points[pbase + 1];
  // clip to [0, dim-1] then truncate (matches astype(int32) on non-negatives)
  const int xi = (int)fminf(fmaxf(px * scale_w, 0.0f), (float)(Wimg - 1));
  const int yi = (int)fminf(fmaxf(py * scale_h, 0.0f), (float)(Himg - 1));

  float* tb = target + (long)b * HW;
  float wsum = 0.0f;
#pragma unroll
  for (int dy = -2; dy <= 2; ++dy) {
#pragma unroll
    for (int dx = -2; dx <= 2; ++dx) {
      const int ny = yi + dy, nx = xi + dx;
      if (ny >= 0 && ny < Himg && nx >= 0 && nx < Wimg) {
        const float wt = expf(-sqrtf((float)(dx * dx + dy * dy)) * 0.5f);
        atomicAdd(tb + (long)ny * Wimg + nx, wt);
        wsum += wt;
      }
    }
  }
  const float s = block_sum(wsum, lds8);  // convergent: after loop, all lanes
  if (threadIdx.x == 0) atomicAdd(&sTgt[b], s);
}

// ---------------------------------------------------------------------------
// Kernel 3: per-image sum of (pred/Sp - target/St)^2 (streaming, float4).
// grid = (BLOCKS_PER_IMG, Bimg), block = 256.
// ---------------------------------------------------------------------------
__global__ void __launch_bounds__(TPB)
k_mse(const float* __restrict__ pred,
      const float* __restrict__ target,
      const float* __restrict__ sPred,
      const float* __restrict__ sTgt,
      float* __restrict__ mseAcc) {
  __shared__ float lds8[TPB / 32];
  const int b = blockIdx.y;
  const float pn = 1.0f / (sPred[b] + 1e-8f);
  const float ts = sTgt[b];
  const float tn = 1.0f / (ts > 0.0f ? ts : 1.0f);

  const long base = (long)b * HW + (long)blockIdx.x * ELEMS_PER_BLOCK;
  float acc = 0.0f;
#pragma unroll
  for (int k = 0; k < 2; ++k) {
    const long e = base + 4L * (threadIdx.x + k * TPB);
    const float4 p = *(const float4*)(pred + e);
    const float4 t = *(const float4*)(target + e);
    const float d0 = p.x * pn - t.x * tn;
    const float d1 = p.y * pn - t.y * tn;
    const float d2 = p.z * pn - t.z * tn;
    const float d3 = p.w * pn - t.w * tn;
    acc += (d0 * d0 + d1 * d1) + (d2 * d2 + d3 * d3);
  }
  const float s = block_sum(acc, lds8);
  if (threadIdx.x == 0) atomicAdd(&mseAcc[b], s);
}

// ---------------------------------------------------------------------------
// Kernel 4: finalize scalar loss.
// ---------------------------------------------------------------------------
__global__ void k_final(const float* __restrict__ sPred,
                        const float* __restrict__ sTgt,
                        const float* __restrict__ mseAcc,
                        const float* __restrict__ cell_area,
                        const float* __restrict__ log_scale,
                        float* __restrict__ out) {
  if (threadIdx.x == 0) {
    const float inv_ca = 1.0f / cell_area[0];
    // multi-scale count loss: every scale yields the same pred_count
    float mean_err = 0.0f;
    for (int b = 0; b < Bimg; ++b)
      mean_err += fabsf(sPred[b] * inv_ca - (float)Npts);
    mean_err *= (1.0f / Bimg);
    const float wsum = (1.0f + 0.5f + 0.25f) / 1.75f;  // == 1
    const float count_loss = wsum * mean_err;

    float spatial = 0.0f;
    for (int b = 0; b < Bimg; ++b)
      spatial += (sTgt[b] > 0.0f) ? mseAcc[b] * (1.0f / (float)HW) : 0.0f;
    spatial *= (1.0f / Bimg);

    const float sv = expf(log_scale[0]);
    const float scale_reg = fmaxf(8.0f - sv, 0.0f) + fmaxf(sv - 64.0f, 0.0f);

    out[0] = 2.0f * count_loss + 0.15f * spatial + 0.5f * scale_reg;
  }
}

// ---------------------------------------------------------------------------
extern "C" void kernel_launch(void* const* d_in, const int* in_sizes, int n_in,
                              void* d_out, int out_size, void* d_ws, size_t ws_size,
                              hipStream_t stream) {
  const float* pred      = (const float*)d_in[0];  // [B,1,H,W]
  const float* points    = (const float*)d_in[1];  // [B,N,2]
  const float* cell_area = (const float*)d_in[2];  // [1]
  const float* log_scale = (const float*)d_in[3];  // [1]
  const int*   H_in_p    = (const int*)d_in[4];    // scalar
  const int*   W_in_p    = (const int*)d_in[5];    // scalar

  // workspace: target [B*HW] floats, then 3*B accumulator floats
  float* target = (float*)d_ws;
  float* acc    = target + (long)Bimg * HW;
  float* sPred  = acc;
  float* sTgt   = acc + Bimg;
  float* mseAcc = acc + 2 * Bimg;

  k_init_acc<<<1, 64, 0, stream>>>(acc);

  dim3 gridStream(BLOCKS_PER_IMG, Bimg);
  k_zero_and_predsum<<<gridStream, TPB, 0, stream>>>(pred, target, sPred);

  dim3 gridPts(Npts / TPB, Bimg);
  k_scatter<<<gridPts, TPB, 0, stream>>>(points, target, sTgt, H_in_p, W_in_p);

  k_mse<<<gridStream, TPB, 0, stream>>>(pred, target, sPred, sTgt, mseAcc);

  k_final<<<1, 32, 0, stream>>>(sPred, sTgt, mseAcc, cell_area, log_scale,
                                (float*)d_out);
}